// SparseGCN_58411555225956
// MI455X (gfx1250) — compile-verified
//
#include <hip/hip_runtime.h>

#define N_NODES   10000
#define N_EDGES   640000
#define IN_FEATS  128
#define H_FEATS   16
#define NUM_CLASSES 8
#define N_TILES   (N_NODES / 16)   // 625 exact

typedef __attribute__((ext_vector_type(2))) float v2f;
typedef __attribute__((ext_vector_type(8))) float v8f;

__device__ __forceinline__ void atomAddF(float* p, float v) {
#if defined(__AMDGCN__)
  unsafeAtomicAdd(p, v);   // lowers to global_atomic_add_f32 / ds_add_f32
#else
  atomicAdd(p, v);
#endif
}

// ---------------------------------------------------------------------------
// Kernel 0: zero Hacc[160000] and s[16]
// ---------------------------------------------------------------------------
__global__ void gcn_zero(float* __restrict__ Hacc, float* __restrict__ s) {
  int tid = blockIdx.x * blockDim.x + threadIdx.x;
  if (tid < N_NODES * H_FEATS) Hacc[tid] = 0.0f;
  if (tid < H_FEATS) s[tid] = 0.0f;
}

// ---------------------------------------------------------------------------
// Kernel 1: XW = features @ W1   (10000x128 @ 128x16) via V_WMMA_F32_16X16X4_F32
// One wave32 per 16-row tile; K accumulated in steps of 4 (32 WMMA ops/tile).
// A 16x4 f32 layout: lanes 0-15 -> M=lane, K={k,k+1}; lanes 16-31 -> K={k+2,k+3}.
// B 4x16 f32 layout mirrors A: lane = N, same K split per half-wave.
// D: VGPR j = row (j + 8*half), col = lane&15.
// ---------------------------------------------------------------------------
__global__ __launch_bounds__(128) void gcn_gemm1_wmma(const float* __restrict__ F,
                                                      const float* __restrict__ W1,
                                                      float* __restrict__ XW) {
  const int lane = threadIdx.x & 31;
  const int wave = threadIdx.x >> 5;
  const int tile = blockIdx.x * 4 + wave;
  if (tile >= N_TILES) return;
  const int r0   = tile * 16;
  const int half = lane >> 4;      // which K-pair this half-wave holds
  const int m    = lane & 15;      // A-row within tile / B,D column

  const float* __restrict__ Arow = F + (size_t)(r0 + m) * IN_FEATS;
  v8f acc = {};
#pragma unroll 8
  for (int k = 0; k < IN_FEATS; k += 4) {
    const int kk = k + half * 2;
    v2f a, b;
    a.x = Arow[kk];
    a.y = Arow[kk + 1];
    b.x = W1[(size_t)kk * H_FEATS + m];
    b.y = W1[(size_t)(kk + 1) * H_FEATS + m];
    acc = __builtin_amdgcn_wmma_f32_16x16x4_f32(false, a, false, b,
                                                (short)0, acc, false, false);
  }
  const int mo = half * 8;
#pragma unroll
  for (int j = 0; j < 8; ++j)
    XW[(size_t)(r0 + mo + j) * H_FEATS + m] = acc[j];
}

// ---------------------------------------------------------------------------
// Kernel 2: SPMM over reduced width: Hacc[row] += (norm[row]*norm[col]) * XW[col]
// 4 threads per edge, each owns a float4 of the 16 features.
// ---------------------------------------------------------------------------
__global__ __launch_bounds__(256) void gcn_spmm1(const int* __restrict__ erow,
                                                 const int* __restrict__ ecol,
                                                 const float* __restrict__ norm,
                                                 const float* __restrict__ XW,
                                                 float* __restrict__ Hacc) {
  const int tid = blockIdx.x * blockDim.x + threadIdx.x;
  const int e = tid >> 2;
  if (e >= N_EDGES) return;
  const int q = tid & 3;
  const int r = erow[e];
  const int c = ecol[e];
  const float val = norm[r] * norm[c];
  const float4 x = ((const float4*)(XW + (size_t)c * H_FEATS))[q];
  float* dst = Hacc + (size_t)r * H_FEATS + q * 4;
  atomAddF(dst + 0, val * x.x);
  atomAddF(dst + 1, val * x.y);
  atomAddF(dst + 2, val * x.z);
  atomAddF(dst + 3, val * x.w);
}

// ---------------------------------------------------------------------------
// Kernel 3: h = relu(Hacc + b1)
// ---------------------------------------------------------------------------
__global__ void gcn_bias_relu(const float* __restrict__ Hacc,
                              const float* __restrict__ b1,
                              float* __restrict__ h) {
  const int tid = blockIdx.x * blockDim.x + threadIdx.x;
  if (tid >= N_NODES * H_FEATS) return;
  const float v = Hacc[tid] + b1[tid & (H_FEATS - 1)];
  h[tid] = v > 0.0f ? v : 0.0f;
}

// ---------------------------------------------------------------------------
// Kernel 4: s[f] = sum_e val_e * h[col_e][f]   (column-sum of spmm2 result)
// Per-thread register accumulation -> LDS ds_add_f32 -> one global atomic/feat.
// ---------------------------------------------------------------------------
__global__ __launch_bounds__(256) void gcn_spmm2_reduce(const int* __restrict__ erow,
                                                        const int* __restrict__ ecol,
                                                        const float* __restrict__ norm,
                                                        const float* __restrict__ h,
                                                        float* __restrict__ s) {
  __shared__ float s_sh[H_FEATS];
  if (threadIdx.x < H_FEATS) s_sh[threadIdx.x] = 0.0f;
  __syncthreads();

  const int tid    = blockIdx.x * blockDim.x + threadIdx.x;
  const int stride = gridDim.x * blockDim.x;
  float4 a0 = {0, 0, 0, 0}, a1 = {0, 0, 0, 0}, a2 = {0, 0, 0, 0}, a3 = {0, 0, 0, 0};
  for (int e = tid; e < N_EDGES; e += stride) {
    const int r = erow[e];
    const int c = ecol[e];
    const float val = norm[r] * norm[c];
    const float4* hc = (const float4*)(h + (size_t)c * H_FEATS);
    float4 h0 = hc[0], h1 = hc[1], h2 = hc[2], h3 = hc[3];
    a0.x += val * h0.x; a0.y += val * h0.y; a0.z += val * h0.z; a0.w += val * h0.w;
    a1.x += val * h1.x; a1.y += val * h1.y; a1.z += val * h1.z; a1.w += val * h1.w;
    a2.x += val * h2.x; a2.y += val * h2.y; a2.z += val * h2.z; a2.w += val * h2.w;
    a3.x += val * h3.x; a3.y += val * h3.y; a3.z += val * h3.z; a3.w += val * h3.w;
  }
  float acc[H_FEATS] = {a0.x, a0.y, a0.z, a0.w, a1.x, a1.y, a1.z, a1.w,
                        a2.x, a2.y, a2.z, a2.w, a3.x, a3.y, a3.z, a3.w};
#pragma unroll
  for (int f = 0; f < H_FEATS; ++f) atomAddF(&s_sh[f], acc[f]);
  __syncthreads();
  if (threadIdx.x < H_FEATS) atomAddF(&s[threadIdx.x], s_sh[threadIdx.x]);
}

// ---------------------------------------------------------------------------
// Kernel 5: out[c] = (1/N) * sum_f s[f] * W2[f][c] + b2[c]
// (mean is linear, so it commutes through @W2 and +b2)
// ---------------------------------------------------------------------------
__global__ void gcn_final(const float* __restrict__ s,
                          const float* __restrict__ W2,
                          const float* __restrict__ b2,
                          float* __restrict__ out) {
  const int c = threadIdx.x;
  if (c >= NUM_CLASSES) return;
  const float invN = 1.0f / (float)N_NODES;
  float acc = b2[c];
#pragma unroll
  for (int f = 0; f < H_FEATS; ++f)
    acc += (s[f] * invN) * W2[f * NUM_CLASSES + c];
  out[c] = acc;
}

// ---------------------------------------------------------------------------
extern "C" void kernel_launch(void* const* d_in, const int* in_sizes, int n_in,
                              void* d_out, int out_size, void* d_ws, size_t ws_size,
                              hipStream_t stream) {
  const float* features = (const float*)d_in[0];
  const float* norm     = (const float*)d_in[1];
  const int*   erow     = (const int*)d_in[2];
  const int*   ecol     = (const int*)d_in[3];
  const float* W1       = (const float*)d_in[4];
  const float* b1       = (const float*)d_in[5];
  const float* W2       = (const float*)d_in[6];
  const float* b2       = (const float*)d_in[7];
  float* out = (float*)d_out;

  float* ws   = (float*)d_ws;
  float* XW   = ws;                        // 160000 floats
  float* Hacc = ws + 160000;               // 160000 floats
  float* h    = ws + 320000;               // 160000 floats
  float* s    = ws + 480000;               // 16 floats

  // 0: zero accumulators
  gcn_zero<<<(N_NODES * H_FEATS + 255) / 256, 256, 0, stream>>>(Hacc, s);
  // 1: XW = X @ W1 (WMMA)
  gcn_gemm1_wmma<<<(N_TILES + 3) / 4, 128, 0, stream>>>(features, W1, XW);
  // 2: Hacc[row] += val * XW[col]
  gcn_spmm1<<<(N_EDGES * 4 + 255) / 256, 256, 0, stream>>>(erow, ecol, norm, XW, Hacc);
  // 3: h = relu(Hacc + b1)
  gcn_bias_relu<<<(N_NODES * H_FEATS + 255) / 256, 256, 0, stream>>>(Hacc, b1, h);
  // 4: s = column-sum of val * h[col]
  gcn_spmm2_reduce<<<256, 256, 0, stream>>>(erow, ecol, norm, h, s);
  // 5: out = s/N @ W2 + b2
  gcn_final<<<1, 32, 0, stream>>>(s, W2, b2, out);
}